// KarpathyTransformerModel_73426760892504
// MI455X (gfx1250) — compile-verified
//
#include <hip/hip_runtime.h>
#include <hip/hip_bf16.h>

// Model dims (compile-time constants from the reference)
#define VOCAB 32000
#define CDIM  64
#define NHEAD 4
#define HS    16
#define TLEN  128
#define BATCH 2
#define HID   256
#define NPE   4
#define NTOK  (BATCH*TLEN)   // 256
#define D4C   256            // 4*C

typedef float v2f  __attribute__((ext_vector_type(2)));
typedef float v8f  __attribute__((ext_vector_type(8)));
typedef unsigned int u32x4 __attribute__((ext_vector_type(4)));
typedef int   i32x4 __attribute__((ext_vector_type(4)));
typedef int   i32x8 __attribute__((ext_vector_type(8)));

// ---------------- block reductions ----------------
template<int NT>
__device__ inline float block_reduce_sum(float v, float* red) {
    int tid = threadIdx.x;
    red[tid] = v; __syncthreads();
    for (int off = NT >> 1; off > 0; off >>= 1) {
        if (tid < off) red[tid] += red[tid + off];
        __syncthreads();
    }
    float r = red[0];
    __syncthreads();
    return r;
}

template<int NT>
__device__ inline float block_reduce_max(float v, float* red) {
    int tid = threadIdx.x;
    red[tid] = v; __syncthreads();
    for (int off = NT >> 1; off > 0; off >>= 1) {
        if (tid < off) red[tid] = fmaxf(red[tid], red[tid + off]);
        __syncthreads();
    }
    float r = red[0];
    __syncthreads();
    return r;
}

__device__ inline float sigmoidf_(float x) { return 1.0f / (1.0f + __expf(-x)); }

// ---------------- TDM: stage a [rows x cols] fp32 row-major matrix into LDS ----------------
// Issued by one wave; caller must s_wait_tensorcnt + barrier afterwards.
__device__ inline void tdm_load_mat_to_lds(unsigned lds_byte_addr, const float* gptr,
                                           unsigned rows, unsigned cols) {
    unsigned long long ga = (unsigned long long)(uintptr_t)gptr;
    u32x4 g0;
    g0[0] = 1u;                                           // count=1 (valid), user mode
    g0[1] = lds_byte_addr;                                // lds_addr
    g0[2] = (unsigned)(ga & 0xFFFFFFFFu);                 // global_addr[31:0]
    g0[3] = (unsigned)((ga >> 32) & 0x01FFFFFFu) | (2u << 30); // addr[56:32] | type=2
    i32x8 g1;
    g1[0] = (int)(2u << 16);                              // wg_mask=0, data_size=4B
    g1[1] = (int)((cols & 0xFFFFu) << 16);                // tensor_dim0 low16
    g1[2] = (int)(((cols >> 16) & 0xFFFFu) | ((rows & 0xFFFFu) << 16)); // dim0 hi | dim1 lo
    g1[3] = (int)(((rows >> 16) & 0xFFFFu) | ((cols & 0xFFFFu) << 16)); // dim1 hi | tile_dim0
    g1[4] = (int)(rows & 0xFFFFu);                        // tile_dim1 | tile_dim2=0
    g1[5] = (int)cols;                                    // tensor_dim0_stride low32
    g1[6] = 0;                                            // stride0 hi | stride1 lo
    g1[7] = 0;                                            // stride1 hi
    i32x4 gz = {0, 0, 0, 0};
#if defined(__clang_major__) && (__clang_major__ >= 23)
    i32x8 gz8 = {0, 0, 0, 0, 0, 0, 0, 0};
    __builtin_amdgcn_tensor_load_to_lds(g0, g1, gz, gz, gz8, 0);
#else
    __builtin_amdgcn_tensor_load_to_lds(g0, g1, gz, gz, 0);
#endif
}

// ---------------- positional embedding: table -> FFN(sigmoid) -> LN ----------------
// grid = NPE*TLEN blocks, 64 threads
__global__ void pe_kernel(const float* __restrict__ tables,
                          const float* __restrict__ W1, const float* __restrict__ b1,
                          const float* __restrict__ W2, const float* __restrict__ b2,
                          const float* __restrict__ g,  const float* __restrict__ bb,
                          float* __restrict__ pos) {
    __shared__ float row[CDIM], hh[CDIM], red[CDIM];
    int p = blockIdx.x >> 7;      // /128
    int t = blockIdx.x & 127;
    int c = threadIdx.x;          // 0..63
    row[c] = tables[(p * TLEN + t) * CDIM + c];
    __syncthreads();
    const float* W1p = W1 + p * CDIM * CDIM;
    float acc = b1[p * CDIM + c];
    for (int k = 0; k < CDIM; k++) acc += row[k] * W1p[k * CDIM + c];
    hh[c] = sigmoidf_(acc);
    __syncthreads();
    const float* W2p = W2 + p * CDIM * CDIM;
    float o = b2[p * CDIM + c];
    for (int k = 0; k < CDIM; k++) o += hh[k] * W2p[k * CDIM + c];
    float m   = block_reduce_sum<CDIM>(o, red) * (1.0f / CDIM);
    float d   = o - m;
    float var = block_reduce_sum<CDIM>(d * d, red) * (1.0f / CDIM);
    float inv = rsqrtf(var + 1e-5f);
    pos[(p * TLEN + t) * CDIM + c] = d * inv * g[p * CDIM + c] + bb[p * CDIM + c];
}

// ---------------- token embedding gather ----------------
// grid = NTOK, 64 threads
__global__ void gather_kernel(const int* __restrict__ idx, const float* __restrict__ emb,
                              float* __restrict__ x) {
    int token = blockIdx.x, c = threadIdx.x;
    x[token * CDIM + c] = emb[idx[token] * CDIM + c];
}

// ---------------- LN1 + x1 concat + V projection ----------------
// grid = NTOK, 64 threads
__global__ void c1_kernel(const float* __restrict__ x, const float* __restrict__ pos_p,
                          const float* __restrict__ g, const float* __restrict__ b_,
                          const float* __restrict__ valW,
                          float* __restrict__ xn, float* __restrict__ x1, float* __restrict__ vv) {
    __shared__ float red[CDIM], xnl[CDIM];
    int token = blockIdx.x;
    int t = token & 127;
    int c = threadIdx.x;
    float xv  = x[token * CDIM + c];
    float m   = block_reduce_sum<CDIM>(xv, red) * (1.0f / CDIM);
    float d   = xv - m;
    float var = block_reduce_sum<CDIM>(d * d, red) * (1.0f / CDIM);
    float xnv = d * rsqrtf(var + 1e-5f) * g[c] + b_[c];
    xnl[c] = xnv;
    xn[token * CDIM + c] = xnv;
    x1[token * 2 * CDIM + c]        = pos_p[t * CDIM + c];
    x1[token * 2 * CDIM + CDIM + c] = xnv;
    __syncthreads();
    int h = c >> 4, s = c & 15;
    const float* vw = valW + h * CDIM * HS;
    float acc = 0.0f;
    for (int k = 0; k < CDIM; k++) acc += xnl[k] * vw[k * HS + s];
    vv[h * (NTOK * HS) + token * HS + s] = acc;   // vv[h][token][s]
}

// ---------------- K/Q projection GEMMs via fp32 WMMA (4-way M-blocked) ----------------
// kq==1: Qp[256tok x 256d]  = x1[256x128] @ W1q[128x256]            (row-major out)
// kq==0: KpT[256d x 256tok] = W1k^T[256x128] @ x1^T[128x256]        (transposed GEMM,
//        coalesced weight reads + coalesced stores, no scatter)
// 512 waves; grid = 64 blocks x 256 threads
__global__ void kq_gemm_kernel(const float* __restrict__ x1, const float* __restrict__ attW1,
                               float* __restrict__ kpT, float* __restrict__ qp) {
    int wid  = blockIdx.x * 8 + (threadIdx.x >> 5);   // 0..511
    int lane = threadIdx.x & 31;
    int gemm = wid >> 6;           // 0..7
    int rem  = wid & 63;
    int mg   = rem & 3;            // M-group of 64 rows (4 tiles of 16)
    int nt   = rem >> 2;           // 0..15
    int h = gemm & 3, kq = gemm >> 2;
    int half = lane >> 4, l16 = lane & 15;
    const float* W = attW1 + h * (D4C * D4C);   // [k:256][d:256]; k rows 0..127 = K, 128..255 = Q
    v8f acc[4];
    for (int i = 0; i < 4; i++) acc[i] = (v8f){0.f,0.f,0.f,0.f,0.f,0.f,0.f,0.f};

    if (kq == 0) {
        // KpT[d][tok] = sum_k W[k][d] * x1[tok][k];  M = d, N = tok
        int n = nt * 16 + l16;                       // token column
        for (int k0 = 0; k0 < 2 * CDIM; k0 += 4) {
            int ka = k0 + 2 * half;
            v2f b = *(const v2f*)&x1[n * (2 * CDIM) + ka];   // B'[ka..ka+1][n]
            for (int mt2 = 0; mt2 < 4; mt2++) {
                int m = mg * 64 + mt2 * 16 + l16;    // d row
                v2f a;
                a.x = W[ka * D4C + m];               // A'[m][ka] = W[ka][m] (lane-coalesced)
                a.y = W[(ka + 1) * D4C + m];
                acc[mt2] = __builtin_amdgcn_wmma_f32_16x16x4_f32(false, a, false, b,
                                                                 (short)0, acc[mt2], false, false);
            }
        }
        for (int mt2 = 0; mt2 < 4; mt2++) {
            int dbase = mg * 64 + mt2 * 16 + 8 * half;
            float* dst = kpT + h * (D4C * NTOK) + dbase * NTOK + nt * 16 + l16;
            for (int r = 0; r < 8; r++) dst[r * NTOK] = acc[mt2][r];   // lane-coalesced
        }
    } else {
        // Qp[tok][d] = sum_k x1[tok][k] * W[128+k][d];  M = tok, N = d
        const float* B = W + (2 * CDIM) * D4C + nt * 16;
        for (int k0 = 0; k0 < 2 * CDIM; k0 += 4) {
            int ka = k0 + 2 * half;
            v2f b;
            b.x = B[ka * D4C + l16];
            b.y = B[(ka + 1) * D4C + l16];
            for (int mt2 = 0; mt2 < 4; mt2++) {
                int arow = mg * 64 + mt2 * 16 + l16;
                v2f a = *(const v2f*)&x1[arow * (2 * CDIM) + ka];
                acc[mt2] = __builtin_amdgcn_wmma_f32_16x16x4_f32(false, a, false, b,
                                                                 (short)0, acc[mt2], false, false);
            }
        }
        for (int mt2 = 0; mt2 < 4; mt2++) {
            int rowbase = mg * 64 + mt2 * 16 + 8 * half;
            float* dst = qp + h * (NTOK * D4C) + rowbase * D4C + nt * 16 + l16;
            for (int r = 0; r < 8; r++) dst[r * D4C] = acc[mt2][r];    // lane-coalesced
        }
    }
}

// ---------------- pairwise additive attention + causal softmax + wei@v ----------------
// grid = (TLEN, NHEAD*BATCH), 128 threads (thread = j)
__global__ void attn_kernel(const float* __restrict__ qp, const float* __restrict__ kpT,
                            const float* __restrict__ vv,
                            const float* __restrict__ b1, const float* __restrict__ w2,
                            const float* __restrict__ b2h, float* __restrict__ o) {
    __shared__ float q[D4C], w2l[D4C], wei[TLEN], red[TLEN];
    int i  = blockIdx.x;
    int hb = blockIdx.y;
    int h = hb >> 1, b = hb & 1;
    int tid = threadIdx.x;
    int token_i = b * TLEN + i;
    for (int d = tid; d < D4C; d += TLEN) {
        q[d]   = qp[h * (NTOK * D4C) + token_i * D4C + d] + b1[h * D4C + d];
        w2l[d] = w2[h * D4C + d];
    }
    __syncthreads();
    int j = tid;
    bool act = (j <= i);
    float s = -3.0e38f;
    if (act) {
        float acc = 0.0f;
        const float* kb = kpT + h * (D4C * NTOK) + b * TLEN + j;
        for (int d = 0; d < D4C; d++) {
            acc += w2l[d] * sigmoidf_(q[d] + kb[d * NTOK]);
        }
        s = (acc + b2h[h]) * 0.125f;   // C^-0.5 = 1/8
    }
    float mx = block_reduce_max<TLEN>(s, red);
    float e = act ? __expf(s - mx) : 0.0f;
    float denom = block_reduce_sum<TLEN>(e, red);
    wei[j] = e / denom;
    __syncthreads();
    if (tid < HS) {
        float acc = 0.0f;
        const float* vb = vv + h * (NTOK * HS) + b * TLEN * HS + tid;
        for (int j2 = 0; j2 <= i; j2++) acc += wei[j2] * vb[j2 * HS];
        o[token_i * CDIM + h * HS + tid] = acc;
    }
}

// ---------------- proj + residual + LN2 + FFN + residual + LN3 ----------------
// grid = NTOK, 256 threads
__global__ void mlp_kernel(const float* __restrict__ o, float* __restrict__ x,
                           const float* __restrict__ projW, const float* __restrict__ projb,
                           const float* __restrict__ g2, const float* __restrict__ bb2,
                           const float* __restrict__ fW1, const float* __restrict__ fb1,
                           const float* __restrict__ fW2, const float* __restrict__ fb2,
                           const float* __restrict__ g3, const float* __restrict__ bb3) {
    __shared__ float ol[CDIM], xl[CDIM], xn2[CDIM], hid[HID], red[HID];
    int token = blockIdx.x;
    int tid = threadIdx.x;
    if (tid < CDIM) { ol[tid] = o[token * CDIM + tid]; xl[tid] = x[token * CDIM + tid]; }
    __syncthreads();
    if (tid < CDIM) {
        float acc = projb[tid];
        for (int k = 0; k < CDIM; k++) acc += ol[k] * projW[k * CDIM + tid];
        xl[tid] += acc;
    }
    __syncthreads();
    // LN2
    float v2 = (tid < CDIM) ? xl[tid] : 0.0f;
    float m2 = block_reduce_sum<HID>(v2, red) * (1.0f / CDIM);
    float d2 = (tid < CDIM) ? (xl[tid] - m2) : 0.0f;
    float var2 = block_reduce_sum<HID>(d2 * d2, red) * (1.0f / CDIM);
    float inv2 = rsqrtf(var2 + 1e-5f);
    if (tid < CDIM) xn2[tid] = d2 * inv2 * g2[tid] + bb2[tid];
    __syncthreads();
    // FFN hidden (256 wide): one thread per hidden unit
    {
        float acc = fb1[tid];
        for (int k = 0; k < CDIM; k++) acc += xn2[k] * fW1[k * HID + tid];
        hid[tid] = sigmoidf_(acc);
    }
    __syncthreads();
    if (tid < CDIM) {
        float f = fb2[tid];
        for (int u = 0; u < HID; u++) f += hid[u] * fW2[u * CDIM + tid];
        xl[tid] += f;
    }
    __syncthreads();
    // LN3 -> write back x
    float v3 = (tid < CDIM) ? xl[tid] : 0.0f;
    float m3 = block_reduce_sum<HID>(v3, red) * (1.0f / CDIM);
    float d3 = (tid < CDIM) ? (xl[tid] - m3) : 0.0f;
    float var3 = block_reduce_sum<HID>(d3 * d3, red) * (1.0f / CDIM);
    float inv3 = rsqrtf(var3 + 1e-5f);
    if (tid < CDIM) x[token * CDIM + tid] = d3 * inv3 * g3[tid] + bb3[tid];
}

// ---------------- LM head GEMM via fp32 WMMA, TDM-staged A, 4-way M-blocked ----------
// out[256x32000] = x[256x64] @ lm_W + lm_b
// wave = 64 rows (4 M-tiles) x 16 cols; 8000 waves; grid = 1000 blocks x 256 threads
__global__ void lm_kernel(const float* __restrict__ x, const float* __restrict__ lmW,
                          const float* __restrict__ lmb, float* __restrict__ out) {
    __shared__ __attribute__((aligned(16))) float As[NTOK * CDIM];   // 64 KB
    // Stage full A (256x64 fp32) into LDS via the Tensor Data Mover (one wave issues).
    if (threadIdx.x < 32) {
        if (threadIdx.x == 0) {
            tdm_load_mat_to_lds((unsigned)(uintptr_t)&As[0], x, NTOK, CDIM);
        }
        __builtin_amdgcn_s_wait_tensorcnt(0);
    }
    __syncthreads();

    int wid  = blockIdx.x * 8 + (threadIdx.x >> 5);
    int lane = threadIdx.x & 31;
    int mg   = wid & 3;          // M-group of 64 rows (4 tiles)
    int nt   = wid >> 2;         // 0..1999
    const float* B = lmW + nt * 16;          // ldb = VOCAB
    int half = lane >> 4, l16 = lane & 15;
    v8f acc[4];
    for (int i = 0; i < 4; i++) acc[i] = (v8f){0.f,0.f,0.f,0.f,0.f,0.f,0.f,0.f};
    for (int k0 = 0; k0 < CDIM; k0 += 4) {
        int ka = k0 + 2 * half;
        v2f b;
        b.x = B[ka * VOCAB + l16];
        b.y = B[(ka + 1) * VOCAB + l16];
        if (k0 + 4 < CDIM) __builtin_prefetch(&B[(ka + 4) * VOCAB + l16], 0, 0);
        for (int mt2 = 0; mt2 < 4; mt2++) {
            int arow = mg * 64 + mt2 * 16 + l16;
            v2f a = *(const v2f*)&As[arow * CDIM + ka];
            acc[mt2] = __builtin_amdgcn_wmma_f32_16x16x4_f32(false, a, false, b,
                                                             (short)0, acc[mt2], false, false);
        }
    }
    int col = nt * 16 + l16;
    float bias = lmb[col];
    for (int mt2 = 0; mt2 < 4; mt2++) {
        float* dst = out + (mg * 64 + mt2 * 16 + 8 * half) * VOCAB + col;
        for (int r = 0; r < 8; r++) dst[r * VOCAB] = acc[mt2][r] + bias;  // lane-coalesced
    }
}

extern "C" void kernel_launch(void* const* d_in, const int* in_sizes, int n_in,
                              void* d_out, int out_size, void* d_ws, size_t ws_size,
                              hipStream_t stream) {
    const int*   idx      = (const int*)  d_in[0];
    const float* tok_emb  = (const float*)d_in[1];
    const float* pe_tab   = (const float*)d_in[2];
    const float* pe_W1    = (const float*)d_in[3];
    const float* pe_b1    = (const float*)d_in[4];
    const float* pe_W2    = (const float*)d_in[5];
    const float* pe_b2    = (const float*)d_in[6];
    const float* pe_g     = (const float*)d_in[7];
    const float* pe_b     = (const float*)d_in[8];
    const float* ln1_g    = (const float*)d_in[9];
    const float* ln1_b    = (const float*)d_in[10];
    const float* att_W1   = (const float*)d_in[11];
    const float* att_b1   = (const float*)d_in[12];
    const float* att_W2   = (const float*)d_in[13];
    const float* att_b2   = (const float*)d_in[14];
    const float* val_W    = (const float*)d_in[15];
    const float* proj_W   = (const float*)d_in[16];
    const float* proj_b   = (const float*)d_in[17];
    const float* ln2_g    = (const float*)d_in[18];
    const float* ln2_b    = (const float*)d_in[19];
    const float* ff_W1    = (const float*)d_in[20];
    const float* ff_b1    = (const float*)d_in[21];
    const float* ff_W2    = (const float*)d_in[22];
    const float* ff_b2    = (const float*)d_in[23];
    const float* ln3_g    = (const float*)d_in[24];
    const float* ln3_b    = (const float*)d_in[25];
    const float* lm_W     = (const float*)d_in[26];
    const float* lm_b     = (const float*)d_in[27];
    float* out = (float*)d_out;

    // workspace layout (floats); total 655360 floats = 2.62 MB
    float* ws  = (float*)d_ws;
    float* pos = ws;            // NPE*T*C        = 32768
    float* x   = ws + 32768;    // NTOK*C         = 16384
    float* xn  = ws + 49152;    // NTOK*C         = 16384
    float* x1  = ws + 65536;    // NTOK*2C        = 32768
    float* vv  = ws + 98304;    // H*NTOK*HS      = 16384
    float* kpT = ws + 114688;   // H*D4C*NTOK     = 262144
    float* qp  = ws + 376832;   // H*NTOK*D4C     = 262144
    float* oa  = ws + 638976;   // NTOK*C         = 16384

    pe_kernel<<<NPE * TLEN, CDIM, 0, stream>>>(pe_tab, pe_W1, pe_b1, pe_W2, pe_b2, pe_g, pe_b, pos);
    gather_kernel<<<NTOK, CDIM, 0, stream>>>(idx, tok_emb, x);
    for (int p = 0; p < NPE; p++) {
        c1_kernel<<<NTOK, CDIM, 0, stream>>>(x, pos + p * TLEN * CDIM, ln1_g, ln1_b, val_W,
                                             xn, x1, vv);
        kq_gemm_kernel<<<64, 256, 0, stream>>>(x1, att_W1, kpT, qp);
        attn_kernel<<<dim3(TLEN, NHEAD * BATCH), TLEN, 0, stream>>>(qp, kpT, vv, att_b1,
                                                                    att_W2, att_b2, oa);
        mlp_kernel<<<NTOK, HID, 0, stream>>>(oa, x, proj_W, proj_b, ln2_g, ln2_b,
                                             ff_W1, ff_b1, ff_W2, ff_b2, ln3_g, ln3_b);
    }
    lm_kernel<<<1000, 256, 0, stream>>>(x, lm_W, lm_b, out);
}